// DiscriminatorMBD_53206054863492
// MI455X (gfx1250) — compile-verified
//
#include <hip/hip_runtime.h>
#include <math.h>

typedef __attribute__((ext_vector_type(2))) float v2f;
typedef __attribute__((ext_vector_type(8))) float v8f;

#define BATCH 512
#define MBD_B 64
#define MBD_C 16
#define FEAT  640
#define CAT   (FEAT + MBD_B)   // 704
#define H1ELEM (BATCH * 20 * 12 * 12)

// ---------------------------------------------------------------------------
// Kernel 1: conv1 (1->20, 5x5, 28->24) + ReLU + 2x2 maxpool -> h1 [512,20,12,12]
// ---------------------------------------------------------------------------
__global__ void k_conv1(const float* __restrict__ x, const float* __restrict__ w,
                        const float* __restrict__ b, float* __restrict__ h1) {
    __shared__ float wsh[20 * 25];
    for (int t = threadIdx.x; t < 20 * 25; t += blockDim.x) wsh[t] = w[t];
    __syncthreads();

    int idx = blockIdx.x * blockDim.x + threadIdx.x;
    if (idx >= H1ELEM) return;
    int px = idx % 12;
    int py = (idx / 12) % 12;
    int oc = (idx / 144) % 20;
    int n  = idx / 2880;

    const float* xi = x + n * 784;
    const float* wk = wsh + oc * 25;
    float bias = b[oc];
    float m = -1e30f;
#pragma unroll
    for (int dy = 0; dy < 2; ++dy)
#pragma unroll
        for (int dx = 0; dx < 2; ++dx) {
            int oy = py * 2 + dy, ox = px * 2 + dx;
            float s = bias;
#pragma unroll
            for (int ky = 0; ky < 5; ++ky)
#pragma unroll
                for (int kx = 0; kx < 5; ++kx)
                    s = fmaf(xi[(oy + ky) * 28 + ox + kx], wk[ky * 5 + kx], s);
            s = fmaxf(s, 0.0f);
            m = fmaxf(m, s);
        }
    h1[idx] = m;
}

// ---------------------------------------------------------------------------
// Kernel 2: conv2 (20->40, 5x5, 12->8) + ReLU + 2x2 maxpool -> feats [512,640]
// Block: 256 threads = 16 images x 16 pooled positions; blockIdx.y = oc.
// ---------------------------------------------------------------------------
__global__ void k_conv2(const float* __restrict__ h1, const float* __restrict__ w,
                        const float* __restrict__ b, float* __restrict__ feats) {
    int oc = blockIdx.y;                 // 0..39
    __shared__ float wsh[20 * 25];       // weights for this oc (2 KB)
    for (int t = threadIdx.x; t < 20 * 25; t += blockDim.x)
        wsh[t] = w[oc * 500 + t];
    __syncthreads();

    int pos  = threadIdx.x & 15;         // 0..15 -> (py,px)
    int nloc = threadIdx.x >> 4;         // 0..15
    int n = blockIdx.x * 16 + nloc;
    int py = pos >> 2, px = pos & 3;

    float bias = b[oc];
    float m = -1e30f;
#pragma unroll
    for (int dy = 0; dy < 2; ++dy)
#pragma unroll
        for (int dx = 0; dx < 2; ++dx) {
            int oy = py * 2 + dy, ox = px * 2 + dx;   // 0..7
            float s = bias;
            for (int ic = 0; ic < 20; ++ic) {
                const float* hp = h1 + ((n * 20 + ic) * 12 + oy) * 12 + ox;
                const float* wk = wsh + ic * 25;
#pragma unroll
                for (int ky = 0; ky < 5; ++ky)
#pragma unroll
                    for (int kx = 0; kx < 5; ++kx)
                        s = fmaf(hp[ky * 12 + kx], wk[ky * 5 + kx], s);
            }
            s = fmaxf(s, 0.0f);
            m = fmaxf(m, s);
        }
    feats[n * FEAT + oc * 16 + py * 4 + px] = m;
}

// ---------------------------------------------------------------------------
// Kernel 3: M = feats[512,640] @ T^T[640,1024]  via V_WMMA_F32_16X16X4_F32
// One wave (32 thr) computes one 16x16 tile of M. grid = (512/16, 1024/16).
// A fragment (16x4 f32): lane<16 holds A[M=lane][k..k+1], lane>=16 holds K=k+2,k+3
// B fragment (4x16 f32): VGPR0 = rows K / K+2 striped over lanes (mirror of C).
// ---------------------------------------------------------------------------
__global__ void k_gemm_mbd(const float* __restrict__ A,   // feats [512,640]
                           const float* __restrict__ Bw,  // T [1024,640]
                           float* __restrict__ Mout) {    // [512,1024]
    int lane = threadIdx.x & 31;
    int half = lane >> 4;     // 0|1
    int l16  = lane & 15;
    int tm = blockIdx.x * 16;
    int tn = blockIdx.y * 16;

    const float* arow = A  + (tm + l16) * FEAT;
    const float* brow = Bw + (tn + l16) * FEAT;

    v8f acc = {};
    for (int k = 0; k < FEAT; k += 4) {
        int ka = k + half * 2;
        v2f a; a.x = arow[ka]; a.y = arow[ka + 1];
        v2f bb; bb.x = brow[ka]; bb.y = brow[ka + 1];
        acc = __builtin_amdgcn_wmma_f32_16x16x4_f32(
            false, a, false, bb, (short)0, acc, false, false);
    }
#pragma unroll
    for (int r = 0; r < 8; ++r)
        Mout[(tm + r + 8 * half) * (MBD_B * MBD_C) + tn + l16] = acc[r];
}

// ---------------------------------------------------------------------------
// Kernel 4: closeness[i,b] = sum_j exp(-sum_c |M[i,b,c]-M[j,b,c]|)
// One thread per (i,b). Lane-consecutive b -> coalesced 2KB/wave reads of M[j].
// ---------------------------------------------------------------------------
__global__ void k_closeness(const float* __restrict__ Mm, float* __restrict__ clos) {
    int idx = blockIdx.x * blockDim.x + threadIdx.x;   // 512*64
    int i = idx >> 6, b = idx & 63;
    const float4* mi4 = (const float4*)(Mm + i * 1024 + b * 16);
    float4 mi0 = mi4[0], mi1 = mi4[1], mi2 = mi4[2], mi3 = mi4[3];

    float sum = 0.0f;
    for (int j = 0; j < BATCH; ++j) {
        const float4* mj4 = (const float4*)(Mm + j * 1024 + b * 16);
        float4 a = mj4[0], c = mj4[1], d = mj4[2], e = mj4[3];
        float dist =
            fabsf(mi0.x - a.x) + fabsf(mi0.y - a.y) + fabsf(mi0.z - a.z) + fabsf(mi0.w - a.w) +
            fabsf(mi1.x - c.x) + fabsf(mi1.y - c.y) + fabsf(mi1.z - c.z) + fabsf(mi1.w - c.w) +
            fabsf(mi2.x - d.x) + fabsf(mi2.y - d.y) + fabsf(mi2.z - d.z) + fabsf(mi2.w - d.w) +
            fabsf(mi3.x - e.x) + fabsf(mi3.y - e.y) + fabsf(mi3.z - e.z) + fabsf(mi3.w - e.w);
        sum += __expf(-dist);
    }
    clos[idx] = sum;
}

// ---------------------------------------------------------------------------
// Kernel 5: fc1: out1[512,100] = relu(cat @ fc1_w^T + b); cat = [feats|clos]
// WMMA f32 16x16x4, N padded 100 -> 112 (zero-filled B, masked store).
// ---------------------------------------------------------------------------
__device__ __forceinline__ float catA(const float* __restrict__ feats,
                                      const float* __restrict__ clos,
                                      int i, int k) {
    return (k < FEAT) ? feats[i * FEAT + k] : clos[i * MBD_B + (k - FEAT)];
}

__global__ void k_fc1(const float* __restrict__ feats, const float* __restrict__ clos,
                      const float* __restrict__ W,   // [100,704]
                      const float* __restrict__ bias,
                      float* __restrict__ out1) {    // [512,100]
    int lane = threadIdx.x & 31;
    int half = lane >> 4;
    int l16  = lane & 15;
    int tm = blockIdx.x * 16;
    int tn = blockIdx.y * 16;
    int row = tm + l16;
    int col = tn + l16;
    bool bvalid = (col < 100);
    const float* brow = W + (bvalid ? col : 0) * CAT;

    v8f acc = {};
    for (int k = 0; k < CAT; k += 4) {
        int ka = k + half * 2;
        v2f a; a.x = catA(feats, clos, row, ka);
               a.y = catA(feats, clos, row, ka + 1);
        v2f bb;
        bb.x = bvalid ? brow[ka]     : 0.0f;
        bb.y = bvalid ? brow[ka + 1] : 0.0f;
        acc = __builtin_amdgcn_wmma_f32_16x16x4_f32(
            false, a, false, bb, (short)0, acc, false, false);
    }
    if (col < 100) {
        float bv = bias[col];
#pragma unroll
        for (int r = 0; r < 8; ++r) {
            int orow = tm + r + 8 * half;
            out1[orow * 100 + col] = fmaxf(acc[r] + bv, 0.0f);
        }
    }
}

// ---------------------------------------------------------------------------
// Kernel 6: fc2 + sigmoid -> out [512,1]
// ---------------------------------------------------------------------------
__global__ void k_fc2(const float* __restrict__ out1, const float* __restrict__ W,
                      const float* __restrict__ bias, float* __restrict__ out) {
    int i = blockIdx.x * blockDim.x + threadIdx.x;
    if (i >= BATCH) return;
    float s = bias[0];
    const float* r = out1 + i * 100;
#pragma unroll 4
    for (int h = 0; h < 100; ++h) s = fmaf(r[h], W[h], s);
    out[i] = 1.0f / (1.0f + __expf(-s));
}

// ---------------------------------------------------------------------------
extern "C" void kernel_launch(void* const* d_in, const int* in_sizes, int n_in,
                              void* d_out, int out_size, void* d_ws, size_t ws_size,
                              hipStream_t stream) {
    const float* x       = (const float*)d_in[0];
    const float* conv1_w = (const float*)d_in[1];
    const float* conv1_b = (const float*)d_in[2];
    const float* conv2_w = (const float*)d_in[3];
    const float* conv2_b = (const float*)d_in[4];
    const float* T       = (const float*)d_in[5];
    const float* fc1_w   = (const float*)d_in[6];
    const float* fc1_b   = (const float*)d_in[7];
    const float* fc2_w   = (const float*)d_in[8];
    const float* fc2_b   = (const float*)d_in[9];
    float* out = (float*)d_out;

    float* ws   = (float*)d_ws;
    float* h1    = ws;                         // 512*20*144   = 1,474,560
    float* feats = h1 + H1ELEM;                // 512*640      =   327,680
    float* Mm    = feats + BATCH * FEAT;       // 512*1024     =   524,288
    float* clos  = Mm + BATCH * 1024;          // 512*64       =    32,768
    float* out1  = clos + BATCH * MBD_B;       // 512*100      =    51,200

    // 1) conv1 + relu + pool
    k_conv1<<<(H1ELEM + 255) / 256, 256, 0, stream>>>(x, conv1_w, conv1_b, h1);
    // 2) conv2 + relu + pool -> feats
    k_conv2<<<dim3(BATCH / 16, 40), 256, 0, stream>>>(h1, conv2_w, conv2_b, feats);
    // 3) M = feats @ T^T   (WMMA f32)
    k_gemm_mbd<<<dim3(BATCH / 16, 1024 / 16), 32, 0, stream>>>(feats, T, Mm);
    // 4) minibatch-discrimination closeness
    k_closeness<<<(BATCH * MBD_B) / 256, 256, 0, stream>>>(Mm, clos);
    // 5) fc1 + relu        (WMMA f32, N padded to 112)
    k_fc1<<<dim3(BATCH / 16, 7), 32, 0, stream>>>(feats, clos, fc1_w, fc1_b, out1);
    // 6) fc2 + sigmoid
    k_fc2<<<(BATCH + 127) / 128, 128, 0, stream>>>(out1, fc2_w, fc2_b, out);
}